// QueryPlanGNN_46540265619522
// MI455X (gfx1250) — compile-verified
//
#include <hip/hip_runtime.h>
#include <hip/hip_bf16.h>

// Problem constants (match reference)
#define GB   128   // batches
#define GN   512   // nodes per graph
#define GK   16    // neighbors
#define GF   32    // input features
#define GH   64    // hidden

typedef __attribute__((ext_vector_type(2))) float v2f;
typedef __attribute__((ext_vector_type(8))) float v8f;
typedef __attribute__((ext_vector_type(4))) int   v4i;

// ---- gfx1250 async global->LDS copy (ASYNCcnt-tracked), with fallback ------
#if defined(__has_builtin)
#  if __has_builtin(__builtin_amdgcn_global_load_async_to_lds_b128)
#    define HAVE_ASYNC_LDS 1
#  endif
#endif
#ifndef HAVE_ASYNC_LDS
#  define HAVE_ASYNC_LDS 0
#endif

#if HAVE_ASYNC_LDS
typedef __attribute__((address_space(1))) v4i gv4i;   // global 16B chunk
typedef __attribute__((address_space(3))) v4i lv4i;   // LDS 16B chunk

__device__ __forceinline__ void async_cp16(const void* g, void* l) {
    __builtin_amdgcn_global_load_async_to_lds_b128(
        (gv4i*)g, (lv4i*)l, /*offset=*/0, /*cpol=*/0);
}
__device__ __forceinline__ void wait_async0() {
#  if __has_builtin(__builtin_amdgcn_s_wait_asynccnt)
    __builtin_amdgcn_s_wait_asynccnt(0);
#  else
    asm volatile("s_wait_asynccnt 0x0" ::: "memory");
#  endif
}
#endif

// D = A(16x4 f32) * B(4x16 f32) + C(16x16 f32), full precision.
// A layout (ISA 7.12.2): lane<16 holds row M=lane, K = k0..k0+1 in 2 VGPRs;
// lane>=16 holds row M=lane-16, K = k0+2..k0+3. B mirrored (N across lanes).
__device__ __forceinline__ v8f wmma_f32_k4(v2f a, v2f b, v8f c) {
    return __builtin_amdgcn_wmma_f32_16x16x4_f32(
        /*neg_a=*/false, a, /*neg_b=*/false, b,
        /*c_mod=*/(short)0, c, /*reuse_a=*/false, /*reuse_b=*/false);
}

// ---------------------------------------------------------------------------
// Encoder: hidden[b,n,h] = relu(nodes[b,n,:] @ enc_w[h,:] + enc_b[h])
// One wave per 16-node tile. GEMM (16 x 32) x (32 x 64) via 8 k-steps x 4 tiles.
// ---------------------------------------------------------------------------
__global__ __launch_bounds__(32)
void gnn_encoder(const float* __restrict__ nodes,
                 const float* __restrict__ enc_w,   // (H, F_IN) row-major
                 const float* __restrict__ enc_b,
                 float* __restrict__ hout)          // (B, N, H)
{
    const int lane  = threadIdx.x & 31;
    const int tile  = blockIdx.x;            // 0 .. B*N/16-1
    const int b     = tile >> 5;             // N/16 = 32 tiles per batch
    const int node0 = (tile & 31) << 4;

    const int m16  = lane & 15;
    const int koff = (lane < 16) ? 0 : 2;

    const float* src = nodes + ((size_t)b * GN + node0) * GF;  // 16 x 32 block
    float* dst = hout + ((size_t)b * GN + node0) * GH;

    for (int t = 0; t < 4; ++t) {
        const int h = t * 16 + m16;          // output column owned by this lane
        v8f acc = {};
#pragma unroll
        for (int k0 = 0; k0 < GF; k0 += 4) {
            v2f a = *(const v2f*)&src[m16 * GF + k0 + koff];
            v2f w = *(const v2f*)&enc_w[h * GF + k0 + koff];
            acc = wmma_f32_k4(a, w, acc);
        }
        const float bb = enc_b[h];
#pragma unroll
        for (int v = 0; v < 8; ++v) {
            const int m = (lane < 16) ? v : v + 8;   // ISA C/D layout
            dst[(size_t)m * GH + h] = fmaxf(acc[v] + bb, 0.0f);
        }
    }
}

// ---------------------------------------------------------------------------
// One message-passing layer:
//   nb_mean = masked mean of gathered neighbor hidden
//   msg     = hidden @ Wself^T + nb_mean @ Wnb^T + bias, zeroed where cnt==0
//   hout    = relu(hidden + msg)
// Block = 64 threads = 2 waves, each wave owns a 16-node tile.
// W (64 x 128) staged in LDS, row stride 132 floats (528 B: 16B-aligned rows
// for async B128 copies; bank = (4h+k) mod 64 -> conflict-free B-slice reads).
// A matrices staged in LDS, row stride 68 (bank-conflict-free A-slice reads).
// ---------------------------------------------------------------------------
#define WSTR 132
#define ASTR 68

__global__ __launch_bounds__(64)
void gnn_layer(const float* __restrict__ hin,
               float* __restrict__ hout,
               const int* __restrict__ adj,       // (B, N, K)
               const float* __restrict__ W,       // (H, 2H) row-major
               const float* __restrict__ bias)    // (H,)
{
    __shared__ float sW[GH * WSTR];                // 33,792 B
    __shared__ float sA[2 * 2 * 16 * ASTR];        // 17,408 B (2 waves x {self, nbmean})
    __shared__ int   sAdj[2 * 16 * GK];            //  2,048 B
    __shared__ float sCnt[2 * 16];                 //    128 B

    const int tid  = threadIdx.x;
    const int lane = tid & 31;
    const int wave = tid >> 5;

    const int gTile = blockIdx.x * 2 + wave;   // 16-node tile id
    const int b     = gTile >> 5;
    const int node0 = (gTile & 31) << 4;

    int* myAdj = sAdj + wave * 16 * GK;
    const int* adjRow = adj + ((size_t)b * GN + node0) * GK;

#if HAVE_ASYNC_LDS
    // Async-stage weights (block-cooperative) and adjacency (per-wave),
    // 16 B per issue, ASYNCcnt-tracked, no VGPR round trip.
    for (int i = tid * 4; i < GH * 2 * GH; i += 64 * 4) {
        const int h = i >> 7;           // /128
        const int k = i & 127;
        async_cp16(W + i, &sW[h * WSTR + k]);
    }
    for (int i = lane * 4; i < 16 * GK; i += 32 * 4)
        async_cp16(adjRow + i, &myAdj[i]);
    wait_async0();
#else
    for (int i = tid; i < GH * 2 * GH; i += 64) {
        const int h = i >> 7;
        const int k = i & 127;
        sW[h * WSTR + k] = W[i];
    }
    for (int i = lane; i < 16 * GK; i += 32) myAdj[i] = adjRow[i];
#endif
    __syncthreads();   // weights + adj visible to whole block

    const float* hb = hin + (size_t)b * GN * GH;
    float* sA1 = sA + wave * (2 * 16 * ASTR);  // self rows
    float* sA2 = sA1 + 16 * ASTR;              // neighbor means

    // Gather + masked mean. Each lane owns the h-pair {2*lane, 2*lane+1}
    // -> one coalesced b64 load per neighbor row (32 lanes cover 256 B row).
    const int h2 = lane * 2;
    for (int m = 0; m < 16; ++m) {
        float s0 = 0.0f, s1 = 0.0f;
        int   c  = 0;
#pragma unroll
        for (int k = 0; k < GK; ++k) {
            const int idx = myAdj[m * GK + k];       // uniform across lanes
            if (idx >= 0) {
                const v2f hv = *(const v2f*)(hb + (size_t)idx * GH + h2);
                s0 += hv.x;
                s1 += hv.y;
                ++c;
            }
        }
        const float inv = 1.0f / fmaxf((float)c, 1.0f);
        v2f mean; mean.x = s0 * inv; mean.y = s1 * inv;
        *(v2f*)&sA2[m * ASTR + h2] = mean;
        *(v2f*)&sA1[m * ASTR + h2] = *(const v2f*)(hb + (size_t)(node0 + m) * GH + h2);
        if (lane == 0) sCnt[wave * 16 + m] = (float)c;
    }
    __syncthreads();

    // GEMM: msg(16x64) = A1(16x64) x Wself^T(64x64) + A2(16x64) x Wnb^T(64x64)
    const int m16  = lane & 15;
    const int koff = (lane < 16) ? 0 : 2;
    float* dst = hout + ((size_t)b * GN + node0) * GH;

    for (int t = 0; t < 4; ++t) {
        const int h = t * 16 + m16;
        v8f acc = {};
#pragma unroll
        for (int k0 = 0; k0 < GH; k0 += 4) {
            // self contribution: B[k][h] = W[h][k]
            v2f a1 = *(const v2f*)&sA1[m16 * ASTR + k0 + koff];
            v2f b1 = *(const v2f*)&sW[h * WSTR + k0 + koff];
            acc = wmma_f32_k4(a1, b1, acc);
            // neighbor contribution: B[k][h] = W[h][H + k]
            v2f a2 = *(const v2f*)&sA2[m16 * ASTR + k0 + koff];
            v2f b2 = *(const v2f*)&sW[h * WSTR + GH + k0 + koff];
            acc = wmma_f32_k4(a2, b2, acc);
        }
        const float bv = bias[h];
#pragma unroll
        for (int v = 0; v < 8; ++v) {
            const int m = (lane < 16) ? v : v + 8;
            float msg = acc[v] + bv;
            msg = (sCnt[wave * 16 + m] > 0.0f) ? msg : 0.0f;
            const float old = sA1[m * ASTR + h];
            dst[(size_t)m * GH + h] = fmaxf(old + msg, 0.0f);
        }
    }
}

// ---------------------------------------------------------------------------
// Head: graph mean-pool -> relu(emb @ out_w1^T + b1) -> @ out_w2^T + b2
// One block per batch, 64 threads (thread h owns hidden channel h).
// ---------------------------------------------------------------------------
__global__ __launch_bounds__(64)
void gnn_head(const float* __restrict__ hidden,   // (B, N, H)
              const float* __restrict__ out_w1,   // (H, H)
              const float* __restrict__ out_b1,
              const float* __restrict__ out_w2,   // (1, H)
              const float* __restrict__ out_b2,
              float* __restrict__ out)            // (B, 1)
{
    __shared__ float se[GH];
    __shared__ float sx[GH];

    const int b = blockIdx.x;
    const int h = threadIdx.x;
    const float* hb = hidden + (size_t)b * GN * GH;

    float s = 0.0f;
    for (int n = 0; n < GN; ++n) s += hb[(size_t)n * GH + h];
    se[h] = s * (1.0f / (float)GN);
    __syncthreads();

    float a = out_b1[h];
#pragma unroll 8
    for (int j = 0; j < GH; ++j) a += se[j] * out_w1[h * GH + j];
    sx[h] = fmaxf(a, 0.0f);
    __syncthreads();

    if (h == 0) {
        float o = out_b2[0];
        for (int j = 0; j < GH; ++j) o += sx[j] * out_w2[j];
        out[b] = o;
    }
}

// ---------------------------------------------------------------------------
extern "C" void kernel_launch(void* const* d_in, const int* in_sizes, int n_in,
                              void* d_out, int out_size, void* d_ws, size_t ws_size,
                              hipStream_t stream) {
    const float* nodes  = (const float*)d_in[0];
    const int*   adj    = (const int*)  d_in[1];
    const float* enc_w  = (const float*)d_in[2];
    const float* enc_b  = (const float*)d_in[3];
    const float* msg_w1 = (const float*)d_in[4];
    const float* msg_b1 = (const float*)d_in[5];
    const float* msg_w2 = (const float*)d_in[6];
    const float* msg_b2 = (const float*)d_in[7];
    const float* msg_w3 = (const float*)d_in[8];
    const float* msg_b3 = (const float*)d_in[9];
    const float* out_w1 = (const float*)d_in[10];
    const float* out_b1 = (const float*)d_in[11];
    const float* out_w2 = (const float*)d_in[12];
    const float* out_b2 = (const float*)d_in[13];

    const size_t hiddenElems = (size_t)GB * GN * GH;   // 4,194,304 floats (16 MB)
    float* h0 = (float*)d_ws;
    float* h1 = h0 + hiddenElems;

    const int nTiles      = GB * GN / 16;   // 4096
    const int layerBlocks = nTiles / 2;     // 2048 (2 waves / block)

    gnn_encoder<<<nTiles, 32, 0, stream>>>(nodes, enc_w, enc_b, h0);
    gnn_layer<<<layerBlocks, 64, 0, stream>>>(h0, h1, adj, msg_w1, msg_b1);
    gnn_layer<<<layerBlocks, 64, 0, stream>>>(h1, h0, adj, msg_w2, msg_b2);
    gnn_layer<<<layerBlocks, 64, 0, stream>>>(h0, h1, adj, msg_w3, msg_b3);
    gnn_head<<<GB, 64, 0, stream>>>(h1, out_w1, out_b1, out_w2, out_b2, (float*)d_out);
}